// ModularityLoss_13434657702449
// MI455X (gfx1250) — compile-verified
//
#include <hip/hip_runtime.h>

// ---------------------------------------------------------------------------
// ModularityLoss, MI455X (gfx1250), wave32, WMMA f32 16x16x4.
//   out = -(T - ||s||^2/(2m)) / (2m)
//   T   = sum(X .* (W @ X))       (only O(N^2 P) term; W streamed ONCE)
//   d   = W @ 1,  2m = sum(W),  s = X^T d
// W is centered by 0.5 in-register (exact identity via column sums of X)
// to keep fp32 WMMA accumulators well-conditioned; final combination in fp64.
// Roofline: W = 256 MB read once -> ~11.5 us floor at 23.3 TB/s HBM.
// Double-buffered LDS staging: next chunk's X loads overlap 128 WMMAs.
// ---------------------------------------------------------------------------

#define NDIM   8192
#define PDIM   128
#define BM     128            // rows of W per workgroup
#define BK     1024           // cols of W per workgroup
#define CBLK   (NDIM / BK)    // 8  col chunks
#define RBLK   (NDIM / BM)    // 64 row blocks
#define NBLK   (RBLK * CBLK)  // 512 workgroups
#define KC     64             // K per LDS stage
#define NCHUNK (BK / KC)      // 16
#define TSTEPS (KC / 4)       // 16 wmma K-steps per stage
#define WS_STRIDE 264         // floats per block: [0..127]=s, 128=T', 129=sumW, [130..257]=colsum(X)
#define CCONST 0.5f

typedef __attribute__((ext_vector_type(2))) float v2f;
typedef __attribute__((ext_vector_type(8))) float v8f;

__global__ __launch_bounds__(256) void mod_main(const float* __restrict__ X,
                                                const float* __restrict__ W,
                                                float* __restrict__ ws) {
  // B fragments pre-swizzled to WMMA layout: [buf][kstep][ntile][lane] = {B[2hl][n], B[2hl+1][n]}
  __shared__ float2 Bfrag[2][TSTEPS][8][32];   // 64 KB, double buffered
  __shared__ float  dchunk[BM];
  __shared__ float  wred[8];

  const int tid  = threadIdx.x;
  const int wave = tid >> 5;
  const int lane = tid & 31;
  const int ll   = lane & 15;   // M (A) / N (B,C) within tile
  const int hl   = lane >> 4;   // K half-select

  const int bid  = blockIdx.x;
  const int rb   = bid >> 3;    // row block  (0..63)
  const int cb   = bid & 7;     // col chunk  (0..7)
  const int row0 = rb * BM;
  const int j0   = cb * BK;

  const int arow = row0 + wave * 16 + ll;               // A-fragment row (M = lane&15)
  const float* __restrict__ Wrow = W + (size_t)arow * NDIM + j0 + 2 * hl;

  // Per-thread staging slice: q = it*256+tid -> k-pair kh = q>>7, column n = q&127
  const int skh = tid >> 7;          // 0 or 1 (adds to 2*it)
  const int sn  = tid & 127;         // column 0..127

  v8f acc[8];
  #pragma unroll
  for (int c = 0; c < 8; ++c) acc[c] = (v8f){};
  float rs = 0.0f;                                      // raw row-sum partial

  float xr[32];                                         // prefetch registers (16 row-pairs)

  // ---- prologue: load + store chunk 0 into buffer 0 ----
  {
    const int kbase = j0;
    #pragma unroll
    for (int it = 0; it < 16; ++it) {
      const int gk = kbase + 2 * (2 * it + skh);
      xr[2 * it]     = X[(size_t)gk * PDIM + sn];
      xr[2 * it + 1] = X[(size_t)(gk + 1) * PDIM + sn];
    }
    #pragma unroll
    for (int it = 0; it < 16; ++it) {
      const int kh = 2 * it + skh;
      Bfrag[0][kh >> 1][sn >> 4][(kh & 1) * 16 + (sn & 15)] =
          make_float2(xr[2 * it], xr[2 * it + 1]);
    }
  }
  __syncthreads();

  for (int kc = 0; kc < NCHUNK; ++kc) {
    // ---- issue next chunk's global loads (latency hidden under WMMAs) ----
    if (kc + 1 < NCHUNK) {
      const int kbase = j0 + (kc + 1) * KC;
      #pragma unroll
      for (int it = 0; it < 16; ++it) {
        const int gk = kbase + 2 * (2 * it + skh);
        xr[2 * it]     = X[(size_t)gk * PDIM + sn];
        xr[2 * it + 1] = X[(size_t)(gk + 1) * PDIM + sn];
      }
    }

    // ---- 16 K-steps of 4: 1 b64 A load + 8 b64 LDS B loads + 8 WMMAs / wave ----
    const int buf = kc & 1;
    const float* __restrict__ wp = Wrow + (size_t)kc * KC;
    #pragma unroll
    for (int t = 0; t < TSTEPS; ++t) {
      const float2 a2 = *(const float2*)(wp + t * 4);
      rs += a2.x + a2.y;                                // true row sum (uncentered)
      v2f a; a.x = a2.x - CCONST; a.y = a2.y - CCONST;  // centered A
      #pragma unroll
      for (int c = 0; c < 8; ++c) {
        const float2 b2 = Bfrag[buf][t][c][lane];
        v2f b; b.x = b2.x; b.y = b2.y;
        acc[c] = __builtin_amdgcn_wmma_f32_16x16x4_f32(
            false, a, false, b, (short)0, acc[c], false, false);
      }
    }

    __syncthreads();   // all waves done reading Bfrag[buf] (and, transitively, buf^1)
    if (kc + 1 < NCHUNK) {
      #pragma unroll
      for (int it = 0; it < 16; ++it) {
        const int kh = 2 * it + skh;
        Bfrag[buf ^ 1][kh >> 1][sn >> 4][(kh & 1) * 16 + (sn & 15)] =
            make_float2(xr[2 * it], xr[2 * it + 1]);
      }
      __syncthreads();
    }
  }

  // ---- T' partial: dot accumulators (C layout: M = v + 8*hl, N = ll) with X tile ----
  float tp = 0.0f;
  #pragma unroll
  for (int c = 0; c < 8; ++c) {
    #pragma unroll
    for (int v = 0; v < 8; ++v) {
      const int m = row0 + wave * 16 + v + 8 * hl;
      tp += acc[c][v] * X[(size_t)m * PDIM + c * 16 + ll];
    }
  }
  #pragma unroll
  for (int off = 16; off > 0; off >>= 1) tp += __shfl_xor(tp, off);

  // ---- per-row chunk sums: lane ll holds K%4 in {2hl, 2hl+1}; fold halves ----
  const float rsf = rs + __shfl_xor(rs, 16);
  if (lane < 16) dchunk[wave * 16 + ll] = rsf;
  if (lane == 0) wred[wave] = tp;
  __syncthreads();

  float* __restrict__ wsb = ws + (size_t)bid * WS_STRIDE;
  if (tid < 128) {
    // s_part[p] = sum_i X[i,p]*dchunk[i] ; colsum_part[p] = sum_i X[i,p]
    float sp = 0.0f, cs = 0.0f;
    const float* __restrict__ xb = X + (size_t)row0 * PDIM + tid;
    for (int i = 0; i < BM; ++i) {
      const float xv = xb[(size_t)i * PDIM];
      sp += xv * dchunk[i];
      cs += xv;
    }
    wsb[tid]       = sp;
    wsb[130 + tid] = cs;
  } else if (tid == 128) {
    float T = 0.0f;
    #pragma unroll
    for (int w2 = 0; w2 < 8; ++w2) T += wred[w2];
    wsb[128] = T;
  } else if (tid == 129) {
    float S = 0.0f;
    for (int i = 0; i < BM; ++i) S += dchunk[i];
    wsb[129] = S;
  }
}

// ---------------------------------------------------------------------------
// Final deterministic reduction in fp64 (the cancellation-sensitive part).
// ---------------------------------------------------------------------------
__global__ __launch_bounds__(256) void mod_reduce(const float* __restrict__ ws,
                                                  float* __restrict__ out) {
  __shared__ double ssq[128];   // s_p^2
  __shared__ double csq[128];   // colsum_p^2
  __shared__ double TS[2];
  const int tid = threadIdx.x;
  if (tid < 128) {
    double s = 0.0, c = 0.0;
    for (int b = 0; b < NBLK; ++b) s += (double)ws[(size_t)b * WS_STRIDE + tid];
    for (int r = 0; r < RBLK; ++r)                       // colsums only from cb==0 blocks
      c += (double)ws[(size_t)(r * CBLK) * WS_STRIDE + 130 + tid];
    ssq[tid] = s * s;
    csq[tid] = c * c;
  } else if (tid == 128) {
    double T = 0.0;
    for (int b = 0; b < NBLK; ++b) T += (double)ws[(size_t)b * WS_STRIDE + 128];
    TS[0] = T;
  } else if (tid == 129) {
    double S = 0.0;
    for (int b = 0; b < NBLK; ++b) S += (double)ws[(size_t)b * WS_STRIDE + 129];
    TS[1] = S;
  }
  __syncthreads();
  if (tid == 0) {
    double sumsq = 0.0, colsq = 0.0;
    for (int p = 0; p < 128; ++p) { sumsq += ssq[p]; colsq += csq[p]; }
    const double S    = TS[1];
    const double m    = fmax(S * 0.5, 1e-8);
    const double twom = 2.0 * m;
    const double T    = TS[0] + (double)CCONST * colsq;  // undo centering exactly
    const double Q    = (T - sumsq / twom) / twom;
    out[0] = (float)(-Q);
  }
}

extern "C" void kernel_launch(void* const* d_in, const int* in_sizes, int n_in,
                              void* d_out, int out_size, void* d_ws, size_t ws_size,
                              hipStream_t stream) {
  const float* X = (const float*)d_in[0];   // (N, P) fp32
  const float* W = (const float*)d_in[1];   // (N, N) fp32
  float* out = (float*)d_out;               // scalar
  float* ws  = (float*)d_ws;                // needs NBLK*WS_STRIDE*4 = 540 KB

  mod_main<<<NBLK, 256, 0, stream>>>(X, W, ws);
  mod_reduce<<<1, 256, 0, stream>>>(ws, out);
}